// FastRayTransformation_18442589569666
// MI455X (gfx1250) — compile-verified
//
#include <hip/hip_runtime.h>
#include <stdint.h>

typedef unsigned int v4u __attribute__((ext_vector_type(4)));
typedef int          v4i __attribute__((ext_vector_type(4)));
typedef int          v8i __attribute__((ext_vector_type(8)));

#ifndef __has_builtin
#define __has_builtin(x) 0
#endif

#if __has_builtin(__builtin_amdgcn_tensor_load_to_lds) && \
    __has_builtin(__builtin_amdgcn_s_wait_tensorcnt)
#define USE_TDM 1
#else
#define USE_TDM 0
#endif

// Problem constants from the reference setup: feat (B,N,C,H,W), out (B,C,V)
#define FRT_B 4
#define FRT_N 6
#define FRT_C 64
#define FRT_H 32
#define FRT_W 88
#define FRT_HW   (FRT_H * FRT_W)        /* 2816    */
#define FRT_CHW  (FRT_C * FRT_HW)       /* 180224  */
#define FRT_NCHW (FRT_N * FRT_CHW)      /* 1081344 */
#define FRT_TILE 256                    /* voxels per block (8 wave32s) */

__global__ __launch_bounds__(FRT_TILE)
void FastRayTransformation_gather_kernel(const float* __restrict__ feat,
                                         const long long* __restrict__ lut,
                                         float* __restrict__ out,
                                         int V)
{
    __shared__ long long s_lut[FRT_TILE * 3];   // 6144 B raw LUT tile
    __shared__ int       s_off[FRT_TILE];       // per-voxel flat offset (-1 = invalid)

    const int tid  = threadIdx.x;
    const int tile = blockIdx.x;
    const int b    = blockIdx.y;

#if USE_TDM
    // ---- Phase 1: TDM async bulk copy of the LUT tile (global -> LDS) ----
    // One wave issues the DMA; TENSORcnt tracks completion.
    if ((tid >> 5) == 0) {
        const uint64_t gaddr = (uint64_t)(uintptr_t)(lut + (size_t)tile * (FRT_TILE * 3));
        const uint32_t laddr = (uint32_t)(uintptr_t)(&s_lut[0]);

        // D# group 0 (CDNA5 ISA):
        //   [1:0] count=1 (valid), [63:32] lds_addr, [120:64] global_addr, [127:126] type=2
        v4u g0;
        g0.x = 1u;
        g0.y = laddr;
        g0.z = (uint32_t)gaddr;
        g0.w = ((uint32_t)(gaddr >> 32) & 0x01FFFFFFu) | (2u << 30);

        // D# group 1 (§8.4): 1 row of 768 8-byte elements.
        //   data_size=3 (8B), tensor_dim0=768, tensor_dim1=1,
        //   tile_dim0=768, tile_dim1=1, tile_dim2=0, tensor_dim0_stride=768
        v8i g1;
        g1[0] = (int)(3u << 16);       // workgroup_mask=0 | data_size=3
        g1[1] = (int)(768u << 16);     // abar_addr=0 | tensor_dim0[15:0]=768
        g1[2] = (int)(1u << 16);       // tensor_dim0[31:16]=0 | tensor_dim1[15:0]=1
        g1[3] = (int)(768u << 16);     // tensor_dim1[31:16]=0 | tile_dim0=768
        g1[4] = 1;                     // tile_dim1=1 | tile_dim2=0
        g1[5] = 768;                   // tensor_dim0_stride[31:0]
        g1[6] = 0;
        g1[7] = 0;

        v4i gz = {0, 0, 0, 0};         // groups 2/3 unused (<=2D tensor)
#if __clang_major__ >= 23
        v8i gz8 = {0, 0, 0, 0, 0, 0, 0, 0};
        __builtin_amdgcn_tensor_load_to_lds(g0, g1, gz, gz, gz8, 0);
#else
        __builtin_amdgcn_tensor_load_to_lds(g0, g1, gz, gz, 0);
#endif
        __builtin_amdgcn_s_wait_tensorcnt(0);
    }
#else
    // Fallback: cooperative direct load of the LUT tile into LDS.
    {
        const long long* lt = lut + (size_t)tile * (FRT_TILE * 3);
        for (int k = tid; k < FRT_TILE * 3; k += FRT_TILE) s_lut[k] = lt[k];
    }
#endif
    __syncthreads();

    // ---- Phase 2a: resolve per-voxel flat offset into feat[b,n,:,v,u] ----
    {
        const long long cam = s_lut[3 * tid + 0];
        const long long uu  = s_lut[3 * tid + 1];
        const long long vv  = s_lut[3 * tid + 2];
        s_off[tid] = (cam < 0) ? -1
                               : (int)(cam * (long long)FRT_CHW + vv * FRT_W + uu);
    }
    __syncthreads();

    // ---- Phase 2b: gather 64 channels (L2-resident feat) and stream out ----
    const int vox = tile * FRT_TILE + tid;
    if (vox >= V) return;

    const int    off   = s_off[tid];
    const float* featb = feat + (size_t)b * FRT_NCHW;
    float*       outp  = out + ((size_t)b * FRT_C) * (size_t)V + vox;

    if (off >= 0) {
        const float* src = featb + off;
#pragma unroll 8
        for (int c = 0; c < FRT_C; ++c) {
            // gather load (regular temporal: keep feat hot in 192MB L2),
            // coalesced non-temporal store (327MB write-once stream).
            __builtin_nontemporal_store(src[(size_t)c * FRT_HW],
                                        outp + (size_t)c * V);
        }
    } else {
#pragma unroll 16
        for (int c = 0; c < FRT_C; ++c) {
            __builtin_nontemporal_store(0.0f, outp + (size_t)c * V);
        }
    }
}

extern "C" void kernel_launch(void* const* d_in, const int* in_sizes, int n_in,
                              void* d_out, int out_size, void* d_ws, size_t ws_size,
                              hipStream_t stream) {
    const float*     feat = (const float*)d_in[0];
    const long long* lut  = (const long long*)d_in[1];
    float*           out  = (float*)d_out;

    const int V = in_sizes[1] / 3;   // 320000 voxels (exact multiple of 256)

    dim3 grid((V + FRT_TILE - 1) / FRT_TILE, FRT_B);
    FastRayTransformation_gather_kernel<<<grid, FRT_TILE, 0, stream>>>(feat, lut, out, V);
}